// ConvBlock_20804821582020
// MI455X (gfx1250) — compile-verified
//
#include <hip/hip_runtime.h>

typedef __attribute__((ext_vector_type(2))) float v2f;
typedef __attribute__((ext_vector_type(8))) float v8f;

#define LRELU_SLOPE 0.2f
#define GAIN_SQRT2  1.41421356237309504880f
#define EPS_DEMOD   1e-8f

#define B_   4
#define NDW_ 512
#define CIN_ 16
#define COUT_ 16
#define H_   1024
#define W_   1024
#define KDIM 144      // CIN * 3 * 3
#define ASTR 146      // padded weight row stride (even -> 8B-aligned pairs)
#define WTILE 128     // pixels per block (8 waves * 16)
#define CSTR 50       // transposed-tile column stride (48 used; even, gcd(50,64)=2)
// K permutation: k' = dx*48 + (cin*3 + dy)
// B element for permuted k' at pixel col c:  xt_t[(c+dx)*CSTR + (cin*3+dy)]
//   -> per lane: base + compile-time offset (dx*CSTR + kk), pair (j=0,1) adjacent.

// ---------------------------------------------------------------------------
// Setup: s[b][i], d[b][o] into workspace; copy w -> head of d_out.
// ---------------------------------------------------------------------------
__global__ __launch_bounds__(256)
void setup_kernel(const float* __restrict__ w,
                  const float* __restrict__ fcw,   // [CIN, NDW]
                  const float* __restrict__ fcb,   // [CIN]
                  const float* __restrict__ cw,    // [COUT, CIN, 3, 3]
                  float* __restrict__ out_w,       // d_out head: B*NDW
                  float* __restrict__ ws_s,        // [B*CIN]
                  float* __restrict__ ws_d) {      // [B*COUT]
    __shared__ float s_sh[64];
    __shared__ float w2_sh[256];
    const int tid = threadIdx.x;

    for (int i = tid; i < B_ * NDW_; i += 256) out_w[i] = w[i];

    if (tid < 64) {
        const int b = tid >> 4, i = tid & 15;
        float acc = 0.f;
        for (int n = 0; n < NDW_; ++n) acc += w[b * NDW_ + n] * fcw[i * NDW_ + n];
        const float s = acc * 0.04419417382415922f /* 512^-0.5 */ + fcb[i];
        s_sh[tid] = s;
        ws_s[tid] = s;
    }
    { // w2[o][i] = sum_k cw^2 / 144
        const int o = tid >> 4, i = tid & 15;
        float acc = 0.f;
        for (int t = 0; t < 9; ++t) {
            const float v = cw[(o * CIN_ + i) * 9 + t];
            acc += v * v;
        }
        w2_sh[tid] = acc * (1.0f / 144.0f);
    }
    __syncthreads();
    if (tid < 64) {
        const int b = tid >> 4, o = tid & 15;
        float acc = 0.f;
        for (int i = 0; i < CIN_; ++i) {
            const float s = s_sh[b * 16 + i];
            acc += s * s * w2_sh[o * 16 + i];
        }
        ws_d[tid] = rsqrtf(acc + EPS_DEMOD);
    }
}

// ---------------------------------------------------------------------------
// Main: implicit-GEMM 3x3 conv via V_WMMA_F32_16X16X4_F32.
// Block = (xb, y0, b): one output row segment of 128 pixels, all 16 couts.
// ---------------------------------------------------------------------------
__global__ __launch_bounds__(256)
void conv_kernel(const float* __restrict__ x,       // [B, CIN, H, W]
                 const float* __restrict__ noise,   // [B, 1, H, W]
                 const float* __restrict__ cw,      // [COUT, CIN, 3, 3]
                 const float* __restrict__ cb,      // [COUT]
                 const float* __restrict__ sn_p,    // scalar
                 const float* __restrict__ ws_s,    // [B*CIN]
                 const float* __restrict__ ws_d,    // [B*COUT]
                 float* __restrict__ out_y) {       // [B, COUT, H, W]
    __shared__ __align__(16) float amod[COUT_ * ASTR]; // modulated weights, K-permuted
    __shared__ __align__(16) float xt[130 * CSTR];     // transposed halo tile [col][cin*3+dy]
    __shared__ float dsc[16];
    __shared__ float bsc[16];

    const int x0  = blockIdx.x * WTILE;
    const int y0  = blockIdx.y;
    const int b   = blockIdx.z;
    const int tid = threadIdx.x;

    // Stage A' = conv_weight * 144^-0.5 * s[b, cin], permuted: k' = dx*48 + cin*3 + dy
    for (int idx = tid; idx < COUT_ * KDIM; idx += 256) {
        const int o  = idx / KDIM;
        const int kp = idx - o * KDIM;
        const int dx = kp / 48;
        const int r  = kp - dx * 48;
        const int ci = r / 3;
        const int dy = r - ci * 3;
        amod[o * ASTR + kp] =
            cw[((o * CIN_ + ci) * 3 + dy) * 3 + dx] * (1.0f / 12.0f) * ws_s[b * 16 + ci];
    }
    // Stage transposed input tile with 1-pixel halo (zero padded at edges).
    // Global reads stay col-fastest (coalesced); LDS writes stride CSTR=50
    // -> 32 distinct banks across a wave (gcd(50,64)=2) -> conflict-free.
    for (int idx = tid; idx < CIN_ * 3 * 130; idx += 256) {
        const int ci  = idx / 390;
        const int rem = idx - ci * 390;
        const int row = rem / 130;
        const int col = rem - row * 130;
        const int gx = x0 - 1 + col;
        const int gy = y0 - 1 + row;
        float v = 0.f;
        if (gx >= 0 && gx < W_ && gy >= 0 && gy < H_)
            v = x[(((size_t)b * CIN_ + ci) * H_ + gy) * W_ + gx];
        xt[col * CSTR + ci * 3 + row] = v;
    }
    if (tid < 16) {
        dsc[tid] = ws_d[b * 16 + tid];
        bsc[tid] = cb[tid];
    }
    __syncthreads();

    const int lane  = tid & 31;
    const int wave  = tid >> 5;
    const int half  = lane >> 4;   // 0: lanes 0-15, 1: lanes 16-31
    const int mn    = lane & 15;   // M for A-frag, N for B/C-frag
    const int wbase = wave * 16;   // this wave's pixel base within the 128-wide tile

    // Fold the runtime 2*half K-offset into per-lane base pointers once:
    // A element (VGPR j, step kb) = amod[mn*ASTR + kb + 2*half + j]
    const float* arow = &amod[mn * ASTR + 2 * half];
    // B element (VGPR j, step kb) = xt[(wbase+mn+dx)*CSTR + kk + 2*half + j]
    const float* bcol = &xt[(wbase + mn) * CSTR + 2 * half];

    v8f acc = {};
#pragma unroll
    for (int ks = 0; ks < 36; ++ks) {
        const int kb = ks * 4;             // permuted-K group base (compile-time)
        const int dx = kb / 48;            // constant per step (groups never straddle)
        const int kk = kb - dx * 48;       // row base within dx segment (even)

        v2f a = *(const v2f*)&arow[kb];                 // ds_load_b64 / 2addr_b64, imm offs
        v2f bf = *(const v2f*)&bcol[dx * CSTR + kk];    // one aligned ds_load_b64, imm off

        acc = __builtin_amdgcn_wmma_f32_16x16x4_f32(
            /*neg_a=*/false, a, /*neg_b=*/false, bf,
            /*c_mod=*/(short)0, acc, /*reuse_a=*/false, /*reuse_b=*/false);
    }

    // Epilogue: demod scale + bias + noise, leaky-ReLU * sqrt(2)
    const float sn   = sn_p[0];
    const int   xpix = x0 + wbase + mn;        // N = mn for both lane halves
    const float nz   = noise[((size_t)b * H_ + y0) * W_ + xpix] * sn;

#pragma unroll
    for (int r = 0; r < 8; ++r) {
        const int M = r + 8 * half;            // C/D: VGPR r holds M=r (lo) / M=8+r (hi)
        float v = acc[r] * dsc[M] + bsc[M] + nz;
        v = GAIN_SQRT2 * (v >= 0.f ? v : LRELU_SLOPE * v);
        out_y[(((size_t)b * COUT_ + M) * H_ + y0) * W_ + xpix] = v;
    }
}

// ---------------------------------------------------------------------------
extern "C" void kernel_launch(void* const* d_in, const int* in_sizes, int n_in,
                              void* d_out, int out_size, void* d_ws, size_t ws_size,
                              hipStream_t stream) {
    const float* w     = (const float*)d_in[0];
    const float* x     = (const float*)d_in[1];
    const float* noise = (const float*)d_in[2];
    const float* fcw   = (const float*)d_in[3];
    const float* fcb   = (const float*)d_in[4];
    const float* cw    = (const float*)d_in[5];
    const float* cb    = (const float*)d_in[6];
    const float* sn    = (const float*)d_in[7];

    float* out  = (float*)d_out;
    float* ws_s = (float*)d_ws;       // 64 floats
    float* ws_d = ws_s + 64;          // 64 floats

    setup_kernel<<<1, 256, 0, stream>>>(w, fcw, fcb, cw, out, ws_s, ws_d);

    dim3 grid(W_ / WTILE, H_, B_);
    conv_kernel<<<grid, 256, 0, stream>>>(x, noise, cw, cb, sn, ws_s, ws_d,
                                          out + B_ * NDW_);
}